// Glove_78073915507326
// MI455X (gfx1250) — compile-verified
//
#include <hip/hip_runtime.h>
#include <hip/hip_bf16.h>
#include <math.h>

typedef __attribute__((ext_vector_type(2))) float v2f;
typedef __attribute__((ext_vector_type(8))) float v8f;

#define EMB   300
#define NF4   75          // 300 floats = 75 float4 per row; rows are 1200 B -> 16B aligned
#define TPB   256
#define NBLK  1024

#define HAVE_WMMA_F32_16X16X4 __has_builtin(__builtin_amdgcn_wmma_f32_16x16x4_f32)

__device__ __forceinline__ float fast_sqrtf(float x) {
#if __has_builtin(__builtin_amdgcn_sqrtf)
    return __builtin_amdgcn_sqrtf(x);   // raw v_sqrt_f32, no IEEE fixup ladder
#else
    return sqrtf(x);
#endif
}

// Main kernel: each wave processes tiles of 16 batch items.
// A-matrix (16x4 f32) layout per CDNA5 ISA: lane l -> row m = l&15;
// lanes 0-15 hold K={0,1}, lanes 16-31 hold K={2,3} (one chunk per VGPR).
// Chunk k = float4 indices j == k (mod 4), so every load is 16B-aligned.
// Bias folded into chunk 0, all chunks scaled by sqrt(w):
//   D[m,n] = sqrt(w_m) * (inner_m + cb_m + ob_m - cooc_m) = sqrt(w_m)*err_m
__global__ __launch_bounds__(TPB) void glove_main(
    const int*   __restrict__ center,  const int*   __restrict__ outside,
    const float* __restrict__ coocs,   const float* __restrict__ wt,
    const float* __restrict__ cemb,    const float* __restrict__ oemb,
    const float* __restrict__ cbias,   const float* __restrict__ obias,
    float* __restrict__ partial, int batch)
{
    const int lane        = threadIdx.x & 31;
    const int wave_in_blk = threadIdx.x >> 5;
    const int wave_id     = blockIdx.x * (TPB / 32) + wave_in_blk;
    const int nwaves      = gridDim.x * (TPB / 32);
    const int m           = lane & 15;             // item-in-tile == A row
    const int k0          = (lane < 16) ? 0 : 2;   // first K-chunk for this lane

    float acc = 0.f;
    const int ntiles = batch >> 4;                 // batch is a multiple of 16

    for (int t = wave_id; t < ntiles; t += nwaves) {
        const int item = (t << 4) + m;
        const int ci = center[item];
        const int oi = outside[item];
        const float4* __restrict__ crow = (const float4*)(cemb + (size_t)ci * EMB);
        const float4* __restrict__ orow = (const float4*)(oemb + (size_t)oi * EMB);

        // Software prefetch of next tile's rows (global_prefetch_b8)
        const int tn = t + nwaves;
        if (tn < ntiles) {
            const int nitem = (tn << 4) + m;
            const int nc = center[nitem];
            const int no = outside[nitem];
            const char* cp = (const char*)(cemb + (size_t)nc * EMB);
            const char* op = (const char*)(oemb + (size_t)no * EMB);
            const int half = (lane < 16) ? 0 : 600;   // two halves of the 1200B row
            __builtin_prefetch(cp + half, 0, 3);
            __builtin_prefetch(op + half, 0, 3);
        }

        // Two chunk dot-products: float4 lanes j == k0, k0+1 (mod 4)
        float a0 = 0.f, a1 = 0.f;
        #pragma unroll 4
        for (int j = k0; j < NF4; j += 4) {
            const float4 c = crow[j], o = orow[j];
            a0 = fmaf(c.x, o.x, a0); a0 = fmaf(c.y, o.y, a0);
            a0 = fmaf(c.z, o.z, a0); a0 = fmaf(c.w, o.w, a0);
        }
        #pragma unroll 4
        for (int j = k0 + 1; j < NF4; j += 4) {
            const float4 c = crow[j], o = orow[j];
            a1 = fmaf(c.x, o.x, a1); a1 = fmaf(c.y, o.y, a1);
            a1 = fmaf(c.z, o.z, a1); a1 = fmaf(c.w, o.w, a1);
        }

        const float s = fast_sqrtf(wt[item]);
        if (lane < 16) {   // fold bias/cooc into chunk 0 (once per item)
            a0 += cbias[ci] + obias[oi] - coocs[item];
        }
        a0 *= s;
        a1 *= s;

#if HAVE_WMMA_F32_16X16X4
        // D[m,n] = sum_k A[m,k] (B = ones) = sqrt(w_m)*err_m, identical across n.
        // EXEC is all-ones here (uniform control flow, full waves).
        v2f A = {a0, a1};
        v2f B = {1.f, 1.f};
        v8f D = {};
        D = __builtin_amdgcn_wmma_f32_16x16x4_f32(
                /*neg_a=*/false, A, /*neg_b=*/false, B,
                /*c_mod=*/(short)0, D, /*reuse_a=*/false, /*reuse_b=*/false);
        // lanes 0..15 hold rows 0..7 in VGPRs 0..7; lanes 16..31 hold rows 8..15.
        // Accumulate each item exactly once: lane 0 -> items 0-7, lane 16 -> items 8-15.
        if ((lane & 15) == 0) {
            #pragma unroll
            for (int r = 0; r < 8; ++r) acc = fmaf(D[r], D[r], acc);
        }
#else
        // Fallback: combine chunk pairs across the two lane halves.
        float tsum = a0 + a1;
        tsum += __shfl_xor(tsum, 16, 32);          // lane m + lane m+16
        if (lane < 16) acc = fmaf(tsum, tsum, acc); // each item once (lanes 0-15)
#endif
    }

    // Wave reduction, then block reduction -> one partial per block
    #pragma unroll
    for (int off = 16; off > 0; off >>= 1) acc += __shfl_down(acc, off, 32);

    __shared__ float red[TPB / 32];
    if (lane == 0) red[wave_in_blk] = acc;
    __syncthreads();
    if (threadIdx.x == 0) {
        float b = 0.f;
        #pragma unroll
        for (int i = 0; i < TPB / 32; ++i) b += red[i];
        partial[blockIdx.x] = b;
    }
}

// Deterministic final reduction of block partials -> d_out[0]
__global__ __launch_bounds__(TPB) void glove_reduce(
    const float* __restrict__ partial, float* __restrict__ out, int n)
{
    float a = 0.f;
    for (int i = threadIdx.x; i < n; i += TPB) a += partial[i];
    #pragma unroll
    for (int off = 16; off > 0; off >>= 1) a += __shfl_down(a, off, 32);

    __shared__ float red[TPB / 32];
    const int lane = threadIdx.x & 31, w = threadIdx.x >> 5;
    if (lane == 0) red[w] = a;
    __syncthreads();
    if (threadIdx.x == 0) {
        float s = 0.f;
        #pragma unroll
        for (int i = 0; i < TPB / 32; ++i) s += red[i];
        out[0] = s;
    }
}

extern "C" void kernel_launch(void* const* d_in, const int* in_sizes, int n_in,
                              void* d_out, int out_size, void* d_ws, size_t ws_size,
                              hipStream_t stream) {
    const int*   center  = (const int*)  d_in[0];
    const int*   outside = (const int*)  d_in[1];
    const float* coocs   = (const float*)d_in[2];
    const float* wt      = (const float*)d_in[3];
    const float* cemb    = (const float*)d_in[4];
    const float* oemb    = (const float*)d_in[5];
    const float* cbias   = (const float*)d_in[6];
    const float* obias   = (const float*)d_in[7];
    const int batch = in_sizes[0];

    int nblk = NBLK;
    const size_t maxblk = ws_size / sizeof(float);
    if ((size_t)nblk > maxblk) nblk = (int)maxblk;
    if (nblk < 1) nblk = 1;

    float* partial = (float*)d_ws;
    glove_main<<<nblk, TPB, 0, stream>>>(center, outside, coocs, wt,
                                         cemb, oemb, cbias, obias,
                                         partial, batch);
    glove_reduce<<<1, TPB, 0, stream>>>(partial, (float*)d_out, nblk);
}